// Newell_Layer_64879775973477
// MI455X (gfx1250) — compile-verified
//
#include <hip/hip_runtime.h>
#include <stdint.h>

#define BLOCK_SIZE 320
#define ROW_DW 5          // 4 data dwords + 1 TDM pad dword per row -> bank-conflict-free
#define MAX_STEPS 300

typedef __attribute__((address_space(1))) int gint_t;
typedef __attribute__((address_space(3))) int lint_t;

typedef unsigned int v4u __attribute__((ext_vector_type(4)));
typedef int          v8i __attribute__((ext_vector_type(8)));
typedef int          v4i __attribute__((ext_vector_type(4)));

#if __has_builtin(__builtin_amdgcn_tensor_load_to_lds)
#define HAVE_TDM 1
#else
#define HAVE_TDM 0
#endif

// Fallback: per-lane async copy of 4 bytes global -> LDS (ASYNCcnt).
__device__ __forceinline__ void async_copy_b32(const float* gptr, float* lptr) {
#if __has_builtin(__builtin_amdgcn_global_load_async_to_lds_b32)
  __builtin_amdgcn_global_load_async_to_lds_b32(
      (gint_t*)gptr, (lint_t*)lptr, /*offset=*/0, /*cpol=*/0);
#else
  uint32_t lds_off = (uint32_t)(uintptr_t)lptr;
  uint64_t ga = (uint64_t)(uintptr_t)gptr;
  asm volatile("global_load_async_to_lds_b32 %0, %1, off"
               :: "v"(lds_off), "v"(ga) : "memory");
#endif
}

__device__ __forceinline__ void wait_async0() {
#if __has_builtin(__builtin_amdgcn_s_wait_asynccnt)
  __builtin_amdgcn_s_wait_asynccnt(0);
#else
  asm volatile("s_wait_asynccnt 0x0" ::: "memory");
#endif
}

#if HAVE_TDM
// One TDM descriptor: 2D tile, 4 elements/row (cols 10..13), `steps` rows,
// row stride 16 elements, 4B data, pad 1 DWORD after every 4 DWORDs in LDS.
__device__ __forceinline__ void tdm_load_tile(const float* g, uint32_t lds_byte,
                                              int steps) {
  uint64_t ga = (uint64_t)(uintptr_t)g;
  v4u g0;
  g0[0] = 1u;                                   // count=1, user descriptor
  g0[1] = lds_byte;                             // lds_addr (bytes)
  g0[2] = (uint32_t)ga;                         // global_addr[31:0]
  g0[3] = (uint32_t)((ga >> 32) & 0x01FFFFFFu)  // global_addr[56:32]
          | (2u << 30);                         // type = 2 ("image")

  v8i g1;
  g1[0] = (2 << 16)      // data_size = 4 bytes
        | (1 << 20)      // pad_enable
        | (1 << 22);     // pad_interval = 1 -> pad after every 4 DWORDs (16B)
                         // pad_amount = 0 -> 1 DWORD; workgroup_mask = 0
  g1[1] = (4 << 16);           // tensor_dim0 = 4 (bits 63:48 = low16)
  g1[2] = (steps << 16);       // tensor_dim0 hi16 = 0 | tensor_dim1 lo16 = steps
  g1[3] = (4 << 16);           // tensor_dim1 hi16 = 0 | tile_dim0 = 4
  g1[4] = steps;               // tile_dim1 = steps | tile_dim2 = 0
  g1[5] = 16;                  // tensor_dim0_stride = 16 elems (low 32 of 48)
  g1[6] = 0;                   // stride0 hi16 | tensor_dim1_stride lo16 (unused)
  g1[7] = 0;

  v4i gz4 = {0, 0, 0, 0};               // groups 2/3: higher dims unused (2D)
  v8i gz8 = {0, 0, 0, 0, 0, 0, 0, 0};   // extra group in 6-arg builtin form
  __builtin_amdgcn_tensor_load_to_lds(g0, g1, gz4, gz4, gz8, /*cpol=*/0);
}
#endif

__global__ __launch_bounds__(BLOCK_SIZE) void newell_forward_kernel(
    const float* __restrict__ x,     // [B, T, F]
    const float* __restrict__ wptr,  // scalar
    float* __restrict__ out,         // [steps, B]
    int B, int T, int F, int steps) {
  __shared__ float S[MAX_STEPS * ROW_DW];  // row t at S[t*5], cols 10..13 in [0..3]
  __shared__ float cbuf[4];                // c_j = td_j * 10
  __shared__ int flags;

  const int b = blockIdx.x;
  const int tid = threadIdx.x;
  const float* xb = x + (size_t)b * T * F;

  if (tid == 0) flags = 0;
  if (tid < 4) {
    // j = tid+1 in reference terms
    const float* xl = xb + (size_t)(T - 1) * F;
    float d = 0.0f;
#pragma unroll
    for (int k = 0; k < 4; ++k)
      if (k <= tid) d += xl[k];  // sum(x_last[:j])
    float td = d * 150.0f / (wptr[0] + xl[5 + tid] * 25.0f);
    cbuf[tid] = td * 10.0f;
  }

#if HAVE_TDM
  // Wave 0 issues one Tensor Data Mover copy for the whole tile, waits on
  // TENSORcnt, then the workgroup barrier publishes the LDS data.
  if (tid < 32) {
    tdm_load_tile(xb + 10, (uint32_t)(uintptr_t)&S[0], steps);
    __builtin_amdgcn_s_wait_tensorcnt(0);
  }
#else
  // Fallback: per-lane CDNA5 async global->LDS copies into the same layout.
  const int total = steps * 4;
  for (int k = tid; k < total; k += BLOCK_SIZE) {
    int t = k >> 2;
    int c = k & 3;
    async_copy_b32(xb + t * F + 10 + c, &S[t * ROW_DW + c]);
  }
  wait_async0();
#endif
  __syncthreads();

  const int i = tid;  // forward step handled by this lane
  float tv0 = 0.f, tv1 = 0.f, tv2 = 0.f, tv3 = 0.f;
  int m = 0;
  if (i < steps) {
    const float fi = (float)i;
    const int hi = steps - 1;  // trunc(i - c) <= i <= steps-1 since c >= 0
    int r0 = (int)(fi - cbuf[0]); r0 = r0 < 0 ? 0 : (r0 > hi ? hi : r0);
    int r1 = (int)(fi - cbuf[1]); r1 = r1 < 0 ? 0 : (r1 > hi ? hi : r1);
    int r2 = (int)(fi - cbuf[2]); r2 = r2 < 0 ? 0 : (r2 > hi ? hi : r2);
    int r3 = (int)(fi - cbuf[3]); r3 = r3 < 0 ? 0 : (r3 > hi ? hi : r3);
    tv0 = S[r0 * ROW_DW + 0];  // stride 5 dwords: gcd(5,64)=1 -> no bank conflicts
    tv1 = S[r1 * ROW_DW + 1];
    tv2 = S[r2 * ROW_DW + 2];
    tv3 = S[r3 * ROW_DW + 3];
    m = (tv0 != 0.f ? 1 : 0) | (tv1 != 0.f ? 2 : 0) |
        (tv2 != 0.f ? 4 : 0) | (tv3 != 0.f ? 8 : 0);
  }
  // wave32 OR-reduction, then one LDS atomic per wave
  int wm = (__any(m & 1) ? 1 : 0) | (__any(m & 2) ? 2 : 0) |
           (__any(m & 4) ? 4 : 0) | (__any(m & 8) ? 8 : 0);
  if ((tid & 31) == 0 && wm) atomicOr(&flags, wm);
  __syncthreads();

  if (i < steps) {
    int f = flags;
    float val = 0.f;  // all-invalid -> zeros (matches reference)
    if (f & 1)      val = tv0;   // argmax(valid) picks the first valid j
    else if (f & 2) val = tv1;
    else if (f & 4) val = tv2;
    else if (f & 8) val = tv3;
    out[(size_t)i * B + b] = val;
  }
}

extern "C" void kernel_launch(void* const* d_in, const int* in_sizes, int n_in,
                              void* d_out, int out_size, void* d_ws, size_t ws_size,
                              hipStream_t stream) {
  // inputs: vi[B], delta_y[B], v_previous[B], x_input[B,T,F], w (scalar)
  const float* x = (const float*)d_in[3];
  const float* w = (const float*)d_in[4];
  float* out = (float*)d_out;

  const int B = in_sizes[0];            // 1024
  const int F = 16;
  const int T = in_sizes[3] / (B * F);  // 2048
  int steps = out_size / B;             // 300
  if (steps > MAX_STEPS) steps = MAX_STEPS;  // LDS tile sized for FORWARD_STEPS=300

  newell_forward_kernel<<<B, BLOCK_SIZE, 0, stream>>>(x, w, out, B, T, F, steps);
}